// MambaBlock_10273561772160
// MI455X (gfx1250) — compile-verified
//
#include <hip/hip_runtime.h>
#include <hip/hip_bf16.h>

typedef __attribute__((ext_vector_type(2)))  float     v2f;
typedef __attribute__((ext_vector_type(8)))  float     v8f;

// ---------------------------------------------------------------------------
// GEMM (NT): C[m][n] = sum_k A[m*lda+k] * B[n*ldb+k]
// Block tile 128(M) x 64(N), 4 waves; each wave a 32x64 tile (2 M-subtiles).
// Double-buffered global->LDS staging (b128 loads), K step 32.
// Requires M%128==0, N%64==0, K%32==0 (true for all uses here).
// ---------------------------------------------------------------------------
__global__ __launch_bounds__(128) void gemm_nt_wmma(
    const float* __restrict__ A, int lda,
    const float* __restrict__ Bm, int ldb,
    float* __restrict__ C, int ldc, int K)
{
    __shared__ float sA[2][128][36];   // pad 36: b128-aligned rows, conflict-free
    __shared__ float sB[2][64][36];
    const int tid  = threadIdx.x;
    const int lane = tid & 31;
    const int wave = tid >> 5;
    const int half = lane >> 4;        // 0: lanes 0-15, 1: lanes 16-31
    const int l16  = lane & 15;
    const int m0   = blockIdx.y * 128;
    const int n0   = blockIdx.x * 64;

    v8f acc[2][4] = {};
    float4 ra[8], rb[4];
    const int nchunks = K >> 5;

    // ---- prologue: load + stage chunk 0 into buffer 0
    #pragma unroll
    for (int it = 0; it < 8; ++it) {
        int i = it * 128 + tid, r = i >> 3, c4 = i & 7;
        ra[it] = *((const float4*)(A + (size_t)(m0 + r) * lda) + c4);
    }
    #pragma unroll
    for (int it = 0; it < 4; ++it) {
        int i = it * 128 + tid, r = i >> 3, c4 = i & 7;
        rb[it] = *((const float4*)(Bm + (size_t)(n0 + r) * ldb) + c4);
    }
    #pragma unroll
    for (int it = 0; it < 8; ++it) {
        int i = it * 128 + tid, r = i >> 3, c4 = i & 7;
        *(float4*)&sA[0][r][c4 * 4] = ra[it];
    }
    #pragma unroll
    for (int it = 0; it < 4; ++it) {
        int i = it * 128 + tid, r = i >> 3, c4 = i & 7;
        *(float4*)&sB[0][r][c4 * 4] = rb[it];
    }

    for (int chunk = 0; chunk < nchunks; ++chunk) {
        __syncthreads();                       // staged chunk visible; prev reads done
        const int  buf  = chunk & 1;
        const bool more = (chunk + 1) < nchunks;

        if (more) {                            // issue next chunk's global loads now
            const int k0 = (chunk + 1) << 5;
            #pragma unroll
            for (int it = 0; it < 8; ++it) {
                int i = it * 128 + tid, r = i >> 3, c4 = i & 7;
                ra[it] = *((const float4*)(A + (size_t)(m0 + r) * lda + k0) + c4);
            }
            #pragma unroll
            for (int it = 0; it < 4; ++it) {
                int i = it * 128 + tid, r = i >> 3, c4 = i & 7;
                rb[it] = *((const float4*)(Bm + (size_t)(n0 + r) * ldb + k0) + c4);
            }
        }

        // ---- compute: 8 K-steps x 8 WMMAs (b frags reused by 2 M-subtiles)
        #pragma unroll
        for (int kk = 0; kk < 32; kk += 4) {
            v2f a0, a1;
            a0.x = sA[buf][wave * 32      + l16][kk + 2 * half];
            a0.y = sA[buf][wave * 32      + l16][kk + 2 * half + 1];
            a1.x = sA[buf][wave * 32 + 16 + l16][kk + 2 * half];
            a1.y = sA[buf][wave * 32 + 16 + l16][kk + 2 * half + 1];
            #pragma unroll
            for (int j = 0; j < 4; ++j) {
                v2f b;
                b.x = sB[buf][j * 16 + l16][kk + 2 * half];
                b.y = sB[buf][j * 16 + l16][kk + 2 * half + 1];
                acc[0][j] = __builtin_amdgcn_wmma_f32_16x16x4_f32(
                    false, a0, false, b, (short)0, acc[0][j], false, false);
                acc[1][j] = __builtin_amdgcn_wmma_f32_16x16x4_f32(
                    false, a1, false, b, (short)0, acc[1][j], false, false);
            }
        }

        if (more) {                            // drain loads into the other buffer
            const int nb = buf ^ 1;
            #pragma unroll
            for (int it = 0; it < 8; ++it) {
                int i = it * 128 + tid, r = i >> 3, c4 = i & 7;
                *(float4*)&sA[nb][r][c4 * 4] = ra[it];
            }
            #pragma unroll
            for (int it = 0; it < 4; ++it) {
                int i = it * 128 + tid, r = i >> 3, c4 = i & 7;
                *(float4*)&sB[nb][r][c4 * 4] = rb[it];
            }
        }
    }

    // C/D layout: VGPR v, lanes0-15 -> row v ; lanes16-31 -> row v+8 ; col = l16
    #pragma unroll
    for (int ms = 0; ms < 2; ++ms)
        #pragma unroll
        for (int j = 0; j < 4; ++j)
            #pragma unroll
            for (int v = 0; v < 8; ++v)
                C[(size_t)(m0 + wave * 32 + ms * 16 + half * 8 + v) * ldc
                  + (n0 + j * 16 + l16)] = acc[ms][j][v];
}

// ---------------------------------------------------------------------------
// Depthwise causal conv(4) + bias + SiLU over the xi half of xz.
// ---------------------------------------------------------------------------
__global__ __launch_bounds__(256) void conv_silu_kernel(
    const float* __restrict__ xz, const float* __restrict__ cw,
    const float* __restrict__ cb, float* __restrict__ xi)
{
    int idx = blockIdx.x * 256 + threadIdx.x;     // over 8192*2048
    int d = idx & 2047;
    int m = idx >> 11;         // b*L + l
    int l = m & 2047;
    const float* base = xz + (size_t)m * 4096 + d;
    float acc = cb[d];
    acc += cw[d * 4 + 3] * base[0];
    if (l >= 1) acc += cw[d * 4 + 2] * base[-4096];
    if (l >= 2) acc += cw[d * 4 + 1] * base[-2 * 4096];
    if (l >= 3) acc += cw[d * 4 + 0] * base[-3 * 4096];
    xi[idx] = acc / (1.0f + __expf(-acc));        // silu
}

// ---------------------------------------------------------------------------
// xd = xi @ W_x^T (N=33). One block per row m; split into dt_in / B / C.
// ---------------------------------------------------------------------------
__global__ __launch_bounds__(256) void xd_kernel(
    const float* __restrict__ xi, const float* __restrict__ Wx,
    float* __restrict__ dtin, float* __restrict__ Bp, float* __restrict__ Cp)
{
    __shared__ float red[256];
    __shared__ float outv[33];
    const int m = blockIdx.x;
    const int tid = threadIdx.x;
    float part[33];
    #pragma unroll
    for (int j = 0; j < 33; ++j) part[j] = 0.0f;
    const float* row = xi + (size_t)m * 2048;
    for (int d = tid; d < 2048; d += 256) {
        float v = row[d];
        #pragma unroll
        for (int j = 0; j < 33; ++j) part[j] += v * Wx[j * 2048 + d];
    }
    for (int j = 0; j < 33; ++j) {
        red[tid] = part[j];
        __syncthreads();
        for (int off = 128; off > 0; off >>= 1) {
            if (tid < off) red[tid] += red[tid + off];
            __syncthreads();
        }
        if (tid == 0) outv[j] = red[0];
        __syncthreads();
    }
    if (tid == 0)       dtin[m] = outv[0];
    if (tid < 16)       Bp[(size_t)m * 16 + tid]        = outv[1 + tid];
    else if (tid < 32)  Cp[(size_t)m * 16 + (tid - 16)] = outv[1 + tid];
}

// ---------------------------------------------------------------------------
// Selective scan. One thread per (b, d) channel; h[16] in VGPRs.
// Block-uniform dt_in/B/C staged via double-buffered LDS, 1 barrier / step.
// ---------------------------------------------------------------------------
__global__ __launch_bounds__(256) void scan_kernel(
    const float* __restrict__ dtin, const float* __restrict__ Bp,
    const float* __restrict__ Cp,   const float* __restrict__ xi,
    const float* __restrict__ Wdt,  const float* __restrict__ bdt,
    const float* __restrict__ A_log, float* __restrict__ y)
{
    __shared__ float sh[2][33];
    const int tid = threadIdx.x;
    const int b   = blockIdx.x >> 3;                 // 8 blocks per batch
    const int d   = (blockIdx.x & 7) * 256 + tid;
    const float wdt = Wdt[d];
    const float bd  = bdt[d];
    float Aa[16], h[16];
    #pragma unroll
    for (int s = 0; s < 16; ++s) {
        Aa[s] = -__expf(fminf(A_log[s], 5.0f));
        h[s]  = 0.0f;
    }
    const size_t mbase = (size_t)b * 2048;
    for (int l = 0; l < 2048; ++l) {
        const size_t m = mbase + l;
        float* buf = sh[l & 1];
        if (tid == 0)       buf[0]   = dtin[m];
        else if (tid <= 16) buf[tid] = Bp[m * 16 + (tid - 1)];
        else if (tid <= 32) buf[tid] = Cp[m * 16 + (tid - 17)];
        __syncthreads();
        float x_t = xi[m * 2048 + d];
        float zv  = buf[0] * wdt + bd;
        float sp  = (zv > 20.0f) ? zv : __logf(1.0f + __expf(zv));
        float dt  = fminf(fmaxf(sp, 1e-4f), 10.0f);
        float yacc = 0.0f;
        #pragma unroll
        for (int s = 0; s < 16; ++s) {
            float dA  = __expf(fmaxf(dt * Aa[s], -20.0f));   // dt*A <= 0 already
            float dBu = fminf(fmaxf(dt * buf[1 + s] * x_t, -10.0f), 10.0f);
            float hn  = fminf(fmaxf(h[s] * dA + dBu, -100.0f), 100.0f);
            h[s] = hn;
            yacc += hn * buf[17 + s];
        }
        y[m * 2048 + d] = yacc;
    }
}

// ---------------------------------------------------------------------------
// u = (layernorm(y)*ln_w + ln_b + D*xi) * silu(z); u overwrites xz[:, :2048].
// ---------------------------------------------------------------------------
__global__ __launch_bounds__(256) void fuse_kernel(
    const float* __restrict__ y, const float* __restrict__ xi,
    float* __restrict__ xz,
    const float* __restrict__ lnw, const float* __restrict__ lnb,
    const float* __restrict__ Dp)
{
    __shared__ float r1[256], r2[256];
    const int m = blockIdx.x;
    const int tid = threadIdx.x;
    const float* yrow = y + (size_t)m * 2048;
    float s = 0.0f, s2 = 0.0f, vals[8];
    #pragma unroll
    for (int i = 0; i < 8; ++i) {
        float v = yrow[tid + i * 256];
        vals[i] = v; s += v; s2 += v * v;
    }
    r1[tid] = s; r2[tid] = s2;
    __syncthreads();
    for (int off = 128; off > 0; off >>= 1) {
        if (tid < off) { r1[tid] += r1[tid + off]; r2[tid] += r2[tid + off]; }
        __syncthreads();
    }
    const float mu   = r1[0] * (1.0f / 2048.0f);
    const float var  = r2[0] * (1.0f / 2048.0f) - mu * mu;
    const float rstd = rsqrtf(var + 1e-5f);
    float* urow = xz + (size_t)m * 4096;
    const float* zrow  = urow + 2048;
    const float* xirow = xi + (size_t)m * 2048;
    #pragma unroll
    for (int i = 0; i < 8; ++i) {
        int dd = tid + i * 256;
        float zv = zrow[dd];
        float sz = zv / (1.0f + __expf(-zv));
        urow[dd] = ((vals[i] - mu) * rstd * lnw[dd] + lnb[dd]
                    + Dp[dd] * xirow[dd]) * sz;
    }
}

// ---------------------------------------------------------------------------
extern "C" void kernel_launch(void* const* d_in, const int* in_sizes, int n_in,
                              void* d_out, int out_size, void* d_ws, size_t ws_size,
                              hipStream_t stream)
{
    const float* x      = (const float*)d_in[0];
    const float* W_in   = (const float*)d_in[1];
    const float* conv_w = (const float*)d_in[2];
    const float* conv_b = (const float*)d_in[3];
    const float* W_x    = (const float*)d_in[4];
    const float* W_dt   = (const float*)d_in[5];
    const float* b_dt   = (const float*)d_in[6];
    const float* A_log  = (const float*)d_in[7];
    const float* D_par  = (const float*)d_in[8];
    const float* W_out  = (const float*)d_in[9];
    const float* ln_w   = (const float*)d_in[10];
    const float* ln_b   = (const float*)d_in[11];
    float* out = (float*)d_out;

    float* ws   = (float*)d_ws;
    float* xz   = ws;                                  // 8192*4096
    float* xi   = xz   + (size_t)8192 * 4096;          // 8192*2048
    float* y    = xi   + (size_t)8192 * 2048;          // 8192*2048
    float* dtin = y    + (size_t)8192 * 2048;          // 8192
    float* Bp   = dtin + 8192;                         // 8192*16
    float* Cp   = Bp   + (size_t)8192 * 16;            // 8192*16

    // 1) xz = x @ W_in^T   (M=8192, N=4096, K=1024)
    gemm_nt_wmma<<<dim3(4096 / 64, 8192 / 128), 128, 0, stream>>>(
        x, 1024, W_in, 1024, xz, 4096, 1024);
    // 2) depthwise conv + bias + silu -> xi
    conv_silu_kernel<<<(8192 * 2048) / 256, 256, 0, stream>>>(
        xz, conv_w, conv_b, xi);
    // 3) xd = xi @ W_x^T -> dt_in, B, C
    xd_kernel<<<8192, 256, 0, stream>>>(xi, W_x, dtin, Bp, Cp);
    // 4) selective scan -> y
    scan_kernel<<<32, 256, 0, stream>>>(dtin, Bp, Cp, xi, W_dt, b_dt, A_log, y);
    // 5) layernorm + gate -> u (written over xz[:, :2048])
    fuse_kernel<<<8192, 256, 0, stream>>>(y, xi, xz, ln_w, ln_b, D_par);
    // 6) out = u @ W_out^T (M=8192, N=1024, K=2048, lda=4096)
    gemm_nt_wmma<<<dim3(1024 / 64, 8192 / 128), 128, 0, stream>>>(
        xz, 4096, W_out, 2048, out, 1024, 2048);
}